// LightGCN_14431090114657
// MI455X (gfx1250) — compile-verified
//
#include <hip/hip_runtime.h>
#include <cstdint>
#include <cstddef>

// ---------------------------------------------------------------------------
// LightGCN propagation for MI455X (gfx1250).
// SpMM is bandwidth/atomic bound (AI ~1.4 flop/byte) -> no WMMA; instead use
// the CDNA5 async global->LDS copy path (ASYNCcnt) to stream edge data,
// double-buffered per wave32, hiding index-fetch latency behind gather+atomics.
// L2 policy: keep x (gather, ~10x reuse) and y (atomic target) resident;
// stream edges (th:TH_LOAD_NT) and accumulator (nontemporal builtins).
// ---------------------------------------------------------------------------

#define WAVES_PER_BLOCK 8
#define CHUNK           32   // edges staged per async batch (one per lane)
#define EPW             128  // edges owned by each wave

typedef float v4f __attribute__((ext_vector_type(4)));

__global__ __launch_bounds__(256) void lgcn_init(const v4f* __restrict__ emb,
                                                 v4f* __restrict__ out,
                                                 v4f* __restrict__ cur,
                                                 v4f* __restrict__ nxt,
                                                 int n4) {
  int i = blockIdx.x * blockDim.x + threadIdx.x;
  if (i < n4) {
    v4f e = __builtin_nontemporal_load(&emb[i]);  // read once ever
    __builtin_nontemporal_store(e, &out[i]);      // acc = x0 (not reused soon)
    cur[i] = e;                                   // layer input: keep RT (hot)
    v4f z = {0.f, 0.f, 0.f, 0.f};
    nxt[i] = z;                                   // spmm target: RT (atomics)
  }
}

// out = (out + nxt) * mul ; zero the consumed buffer so it can be the next
// spmm scatter target.
__global__ __launch_bounds__(256) void lgcn_acc(v4f* __restrict__ out,
                                                const v4f* __restrict__ nxt,
                                                v4f* __restrict__ curz,
                                                int n4, float mul) {
  int i = blockIdx.x * blockDim.x + threadIdx.x;
  if (i < n4) {
    v4f o = __builtin_nontemporal_load(&out[i]);  // acc: low reuse -> NT
    v4f x = nxt[i];                               // becomes next layer's x: RT
    o = (o + x) * mul;
    __builtin_nontemporal_store(o, &out[i]);
    v4f z = {0.f, 0.f, 0.f, 0.f};
    curz[i] = z;                                  // next atomic target: RT
  }
}

// COO SpMM: y[row] += val * x[col], one wave32 per edge, 2 features per lane.
// Edge triples are streamed into LDS with gfx1250 async copies (non-temporal,
// once-per-layer streams), double buffered, waits via s_wait_asynccnt
// (async loads complete in order).
__global__ __launch_bounds__(256) void lgcn_spmm(
    const long long* __restrict__ rowp,
    const long long* __restrict__ colp,
    const float*     __restrict__ valp,
    const float*     __restrict__ x,
    float*           __restrict__ y,
    int E) {
  __shared__ long long s_row[WAVES_PER_BLOCK][2][CHUNK];
  __shared__ long long s_col[WAVES_PER_BLOCK][2][CHUNK];
  __shared__ float     s_val[WAVES_PER_BLOCK][2][CHUNK];

  const int lane = threadIdx.x & 31;
  const int wiw  = threadIdx.x >> 5;  // wave index within workgroup
  const long long wave  = (long long)blockIdx.x * WAVES_PER_BLOCK + wiw;
  const long long start = wave * (long long)EPW;
  if (start >= (long long)E) return;
  const long long end =
      (start + EPW < (long long)E) ? (start + EPW) : (long long)E;

  // Issue one CHUNK of edge data into LDS buffer `buf` (3 async ops).
  auto issue = [&](long long e0, int buf) {
    long long e = e0 + lane;
    if (e >= (long long)E) e = (long long)E - 1;  // clamp: benign duplicate
    unsigned g8 = (unsigned)(e << 3);  // byte offset into i64 arrays
    unsigned g4 = (unsigned)(e << 2);  // byte offset into f32 array
    unsigned lc = (unsigned)(size_t)&s_col[wiw][buf][lane];
    unsigned lr = (unsigned)(size_t)&s_row[wiw][buf][lane];
    unsigned lv = (unsigned)(size_t)&s_val[wiw][buf][lane];
    asm volatile("global_load_async_to_lds_b64 %0, %1, %2 th:TH_LOAD_NT"
                 :: "v"(lc), "v"(g8), "s"(colp) : "memory");
    asm volatile("global_load_async_to_lds_b64 %0, %1, %2 th:TH_LOAD_NT"
                 :: "v"(lr), "v"(g8), "s"(rowp) : "memory");
    asm volatile("global_load_async_to_lds_b32 %0, %1, %2 th:TH_LOAD_NT"
                 :: "v"(lv), "v"(g4), "s"(valp) : "memory");
  };

  int buf = 0;
  issue(start, 0);
  for (long long e0 = start; e0 < end; e0 += CHUNK) {
    const bool more = (e0 + CHUNK) < end;
    if (more) {
      issue(e0 + CHUNK, buf ^ 1);
      // 3 newly-issued ops outstanding; async loads retire in order, so
      // cnt<=3 guarantees the current buffer's 3 ops are complete.
      asm volatile("s_wait_asynccnt 0x3" ::: "memory");
    } else {
      asm volatile("s_wait_asynccnt 0x0" ::: "memory");
    }

    const int cnt = (int)(((end - e0) < (long long)CHUNK) ? (end - e0)
                                                          : (long long)CHUNK);
    const int fo = lane * 2;  // this lane's feature pair
    for (int i = 0; i < cnt; ++i) {
      // same-address reads -> broadcast to all 32 lanes (volatile: the only
      // writer the compiler can see is the asm, so loads must not be elided)
      long long c = *(volatile long long*)&s_col[wiw][buf][i];
      long long r = *(volatile long long*)&s_row[wiw][buf][i];
      float     v = *(volatile float*)&s_val[wiw][buf][i];
      const float2 xv = *(const float2*)(x + c * 64 + fo);  // coalesced 256B
      atomicAdd(y + r * 64 + fo,     v * xv.x);             // L2 atomic f32
      atomicAdd(y + r * 64 + fo + 1, v * xv.y);
    }
    buf ^= 1;
  }
}

extern "C" void kernel_launch(void* const* d_in, const int* in_sizes, int n_in,
                              void* d_out, int out_size, void* d_ws, size_t ws_size,
                              hipStream_t stream) {
  const float*     emb = (const float*)d_in[0];      // [N,64] fp32
  const long long* row = (const long long*)d_in[1];  // [E] i64
  const long long* col = (const long long*)d_in[2];  // [E] i64
  const float*     val = (const float*)d_in[3];      // [E] fp32
  (void)n_in; (void)ws_size; (void)out_size;

  const int N = in_sizes[0] / 64;
  const int E = in_sizes[1];
  const long long nElem = (long long)N * 64;
  const int n4 = (int)(nElem / 4);

  float* outp = (float*)d_out;
  float* cur  = (float*)d_ws;          // 102.4 MB
  float* nxt  = cur + nElem;           // 102.4 MB

  const int ewBlocks = (n4 + 255) / 256;
  lgcn_init<<<ewBlocks, 256, 0, stream>>>(
      (const v4f*)emb, (v4f*)outp, (v4f*)cur, (v4f*)nxt, n4);

  const long long waves = ((long long)E + EPW - 1) / EPW;
  const int spmmBlocks  = (int)((waves + WAVES_PER_BLOCK - 1) / WAVES_PER_BLOCK);

  for (int layer = 0; layer < 3; ++layer) {
    lgcn_spmm<<<spmmBlocks, 256, 0, stream>>>(row, col, val, cur, nxt, E);
    const float mul = (layer == 2) ? 0.25f : 1.0f;  // fuse final /4
    lgcn_acc<<<ewBlocks, 256, 0, stream>>>(
        (v4f*)outp, (const v4f*)nxt, (v4f*)cur, n4, mul);
    float* t = cur; cur = nxt; nxt = t;
  }
}